// MemoryCell_16217796510025
// MI455X (gfx1250) — compile-verified
//
#include <hip/hip_runtime.h>
#include <math.h>

#define H   1024
#define BSZ 1024   // batch (enc rows); == H (the reference broadcast quirk relies on this)
#define T   128
#define NB  5

typedef __attribute__((ext_vector_type(2))) float v2f;
typedef __attribute__((ext_vector_type(8))) float v8f;

// ---------------------------------------------------------------------------
// Stage 1: ws_a[n][j] = h[n]·Uw[j] + keys[n]·Vw[j]      (5 x 1024 outputs)
// one wave32 per output, lane-strided K, shfl reduction
// ---------------------------------------------------------------------------
__global__ __launch_bounds__(256)
void stage_a_kernel(const float* __restrict__ states,
                    const float* __restrict__ Uw,
                    const float* __restrict__ Vw,
                    const float* __restrict__ keys,
                    float* __restrict__ ws_a) {
    const int lane = threadIdx.x & 31;
    const int wid  = (blockIdx.x * blockDim.x + threadIdx.x) >> 5;
    const int n = wid >> 10;
    const int j = wid & 1023;
    if (n >= NB) return;
    const float* hrow = states + n * H;
    const float* krow = keys   + n * H;
    const float* urow = Uw + (size_t)j * H;
    const float* vrow = Vw + (size_t)j * H;
    float acc = 0.f;
    for (int k = lane; k < H; k += 32)
        acc = fmaf(hrow[k], urow[k], fmaf(krow[k], vrow[k], acc));
    for (int off = 16; off; off >>= 1) acc += __shfl_xor(acc, off, 32);
    if (lane == 0) ws_a[n * H + j] = acc;
}

// ---------------------------------------------------------------------------
// Stage 2: ws_gate[n][b] = sigmoid(enc[b]·(h[n]+keys[n]))   (5 x 1024 outputs)
// ---------------------------------------------------------------------------
__global__ __launch_bounds__(256)
void stage_gate_kernel(const float* __restrict__ features,
                       const float* __restrict__ states,
                       const float* __restrict__ keys,
                       float* __restrict__ ws_gate) {
    const int lane = threadIdx.x & 31;
    const int wid  = (blockIdx.x * blockDim.x + threadIdx.x) >> 5;
    const int n = wid >> 10;
    const int b = wid & 1023;
    if (n >= NB) return;
    const float* enc  = features + (size_t)b * (T * H);   // features[b, 0, :]
    const float* hrow = states + n * H;
    const float* krow = keys   + n * H;
    float acc = 0.f;
    for (int k = lane; k < H; k += 32)
        acc = fmaf(enc[k], hrow[k] + krow[k], acc);
    for (int off = 16; off; off >>= 1) acc += __shfl_xor(acc, off, 32);
    if (lane == 0) ws_gate[n * H + b] = 1.f / (1.f + __expf(-acc));
}

// ---------------------------------------------------------------------------
// Stage 3: P = enc @ Ww.T via V_WMMA_F32_16X16X4_F32, fused epilogue:
//   pre  = ws_a[n,j] + P[b,j]
//   cand = pre >= 0 ? pre : prelu_a[j]*pre
//   val  = h[n,j] + ws_gate[n][j]*cand     (gate indexed by COLUMN j — ref quirk)
//   out[(n*1024+b)*1024+j] = val < 0 ? -1 : +1   (val==0 -> 0.1 -> +1)
//
// Block: 256 threads = 8 waves, tile 64(M) x 128(N), K-slab = 16 via LDS.
// Wave (waveM in 0..1, waveN in 0..3) owns a 32x32 region = 2x2 WMMA tiles.
// Register-level software pipeline: global loads for slab i+1 are issued
// before the compute of slab i, so s_wait_loadcnt lands after 16 WMMAs.
// ---------------------------------------------------------------------------
#define LDA 17    // 64 rows x 16 K, padded (17 coprime with 64 banks)
#define LDB 136   // 16 K-rows x 128 cols, padded (2*136 % 64 == 16 -> halves disjoint)

__global__ __launch_bounds__(256)
void gemm_sign_kernel(const float* __restrict__ features,
                      const float* __restrict__ Ww,
                      const float* __restrict__ states,
                      const float* __restrict__ prelu_a,
                      const float* __restrict__ ws_a,
                      const float* __restrict__ ws_gate,
                      float* __restrict__ out) {
    __shared__ float ldsA[64 * LDA];
    __shared__ float ldsB[16 * LDB];

    const int tid   = threadIdx.x;
    const int lane  = tid & 31;
    const int wid   = tid >> 5;
    const int waveM = wid & 1;     // 2 wave-rows of 32
    const int waveN = wid >> 1;    // 4 wave-cols of 32
    const int blockM = blockIdx.x * 64;
    const int blockN = blockIdx.y * 128;

    v8f acc[2][2];
    const v8f vzero = {0.f, 0.f, 0.f, 0.f, 0.f, 0.f, 0.f, 0.f};
    acc[0][0] = vzero; acc[0][1] = vzero; acc[1][0] = vzero; acc[1][1] = vzero;

    // per-thread staging coordinates
    const int arow = tid >> 2;          // 0..63
    const int akq  = (tid & 3) << 2;    // 0,4,8,12
    const int brow = tid >> 1;          // 0..127 (row of Ww inside block tile)
    const int bkq  = (tid & 1) << 3;    // 0 or 8

    const float* Aptr = features + (size_t)(blockM + arow) * (T * H) + akq; // enc[row, k]
    const float* Bptr = Ww + (size_t)(blockN + brow) * H + bkq;             // Ww[col, k]

    const int mrow = waveM * 32 + (lane & 15);   // A row within block tile (tile 0)
    const int ncol = waveN * 32 + (lane & 15);   // B col within block tile (tile 0)
    const int khi  = (lane >> 4) << 1;           // lanes 16-31 take K = k4+2, k4+3

    // ---- software pipeline: preload slab 0 into registers ----
    float4 av  = *(const float4*)(Aptr);
    float4 bv0 = *(const float4*)(Bptr);
    float4 bv1 = *(const float4*)(Bptr + 4);

    for (int k0 = 0; k0 < H; k0 += 16) {
        __syncthreads();                       // previous compute done; LDS reusable
        ldsA[arow * LDA + akq + 0] = av.x;
        ldsA[arow * LDA + akq + 1] = av.y;
        ldsA[arow * LDA + akq + 2] = av.z;
        ldsA[arow * LDA + akq + 3] = av.w;
        ldsB[(bkq + 0) * LDB + brow] = bv0.x;
        ldsB[(bkq + 1) * LDB + brow] = bv0.y;
        ldsB[(bkq + 2) * LDB + brow] = bv0.z;
        ldsB[(bkq + 3) * LDB + brow] = bv0.w;
        ldsB[(bkq + 4) * LDB + brow] = bv1.x;
        ldsB[(bkq + 5) * LDB + brow] = bv1.y;
        ldsB[(bkq + 6) * LDB + brow] = bv1.z;
        ldsB[(bkq + 7) * LDB + brow] = bv1.w;
        __syncthreads();

        // issue global loads for the NEXT slab; the s_wait_loadcnt for these
        // lands at the top of the next iteration, hidden behind the WMMAs below
        if (k0 + 16 < H) {
            av  = *(const float4*)(Aptr + k0 + 16);
            bv0 = *(const float4*)(Bptr + k0 + 16);
            bv1 = *(const float4*)(Bptr + k0 + 20);
        }
        if (k0 + 32 < H) {                     // L2 hint one stage further ahead
            __builtin_prefetch((const void*)(Aptr + k0 + 32), 0, 1);
            __builtin_prefetch((const void*)(Bptr + k0 + 32), 0, 1);
        }

#pragma unroll
        for (int k4 = 0; k4 < 16; k4 += 4) {
            const int kA = k4 + khi;
            v2f aF0, aF1, bF0, bF1;
            aF0.x = ldsA[(mrow     ) * LDA + kA];
            aF0.y = ldsA[(mrow     ) * LDA + kA + 1];
            aF1.x = ldsA[(mrow + 16) * LDA + kA];
            aF1.y = ldsA[(mrow + 16) * LDA + kA + 1];
            bF0.x = ldsB[(kA    ) * LDB + ncol];
            bF0.y = ldsB[(kA + 1) * LDB + ncol];
            bF1.x = ldsB[(kA    ) * LDB + ncol + 16];
            bF1.y = ldsB[(kA + 1) * LDB + ncol + 16];
            acc[0][0] = __builtin_amdgcn_wmma_f32_16x16x4_f32(false, aF0, false, bF0,
                                                              (short)0, acc[0][0], false, false);
            acc[0][1] = __builtin_amdgcn_wmma_f32_16x16x4_f32(false, aF0, false, bF1,
                                                              (short)0, acc[0][1], false, false);
            acc[1][0] = __builtin_amdgcn_wmma_f32_16x16x4_f32(false, aF1, false, bF0,
                                                              (short)0, acc[1][0], false, false);
            acc[1][1] = __builtin_amdgcn_wmma_f32_16x16x4_f32(false, aF1, false, bF1,
                                                              (short)0, acc[1][1], false, false);
        }
    }

    // ---- fused epilogue ----
    const int colHalf = lane & 15;
    const int rowAdd  = (lane >> 4) << 3;   // C/D layout: VGPR r -> M=r (lo half) / r+8 (hi half)

#pragma unroll
    for (int tn = 0; tn < 2; ++tn) {
        const int gc = blockN + waveN * 32 + tn * 16 + colHalf;
        const float pa = prelu_a[gc];
        float avn[NB], hvn[NB], gvn[NB];
#pragma unroll
        for (int n = 0; n < NB; ++n) {
            avn[n] = ws_a[n * H + gc];
            hvn[n] = states[n * H + gc];
            gvn[n] = ws_gate[n * H + gc];   // gate[j, n] with j == column (ref quirk)
        }
#pragma unroll
        for (int tm = 0; tm < 2; ++tm) {
            const int grBase = blockM + waveM * 32 + tm * 16 + rowAdd;
            v8f c = acc[tm][tn];
#pragma unroll
            for (int r = 0; r < 8; ++r) {
                const float P = c[r];
#pragma unroll
                for (int n = 0; n < NB; ++n) {
                    float pre  = avn[n] + P;
                    float cand = (pre >= 0.f) ? pre : pa * pre;
                    float val  = fmaf(gvn[n], cand, hvn[n]);
                    // val==0 -> 0.1 -> +1 ; else sign(val). -0.0 also -> +1.
                    out[(size_t)(n * BSZ + grBase + r) * H + gc] = (val < 0.f) ? -1.f : 1.f;
                }
            }
        }
    }
}

// ---------------------------------------------------------------------------
extern "C" void kernel_launch(void* const* d_in, const int* in_sizes, int n_in,
                              void* d_out, int out_size, void* d_ws, size_t ws_size,
                              hipStream_t stream) {
    const float* features = (const float*)d_in[0];
    const float* states   = (const float*)d_in[1];
    const float* Uw       = (const float*)d_in[2];
    const float* Vw       = (const float*)d_in[3];
    const float* Ww       = (const float*)d_in[4];
    const float* keys     = (const float*)d_in[5];
    const float* prelu_a  = (const float*)d_in[6];
    float* out     = (float*)d_out;
    float* ws_a    = (float*)d_ws;           // NB*H floats
    float* ws_gate = ws_a + NB * H;          // NB*H floats  (40 KB total)

    stage_a_kernel<<<(NB * H) / 8, 256, 0, stream>>>(states, Uw, Vw, keys, ws_a);
    stage_gate_kernel<<<(NB * BSZ) / 8, 256, 0, stream>>>(features, states, keys, ws_gate);
    dim3 grid(BSZ / 64, H / 128);
    gemm_sign_kernel<<<grid, 256, 0, stream>>>(features, Ww, states, prelu_a,
                                               ws_a, ws_gate, out);
}